// ncc_loss_32590211842412
// MI455X (gfx1250) — compile-verified
//
#include <hip/hip_runtime.h>

// NCC loss, fused single pass:
//   5 box sums via V_WMMA_F32_16X16X4_F32 (separable banded-ones matmuls),
//   per-pixel cc, wave reduction, f64 atomic accumulate, finalize to mean.

typedef __attribute__((ext_vector_type(2))) float v2f;
typedef __attribute__((ext_vector_type(8))) float v8f;

#define H_IMG 1024
#define W_IMG 1024
#define N_IMG 8
#define TILES_X 64
#define TILES_Y 64
#define TILES_PER_IMG (TILES_X * TILES_Y)
#define TOTAL_TILES (N_IMG * TILES_PER_IMG)
#define WAVES_PER_BLOCK 4
#define TSTRIDE 25   // 24-wide tile rows, +1 pad vs 64 LDS banks
#define HSTRIDE 17   // 16-wide H rows, +1 pad

__device__ __forceinline__ v8f wmma_f32(v2f a, v2f b, v8f c) {
  // 8 args: (neg_a, A, neg_b, B, c_mod, C, reuse_a, reuse_b)
  return __builtin_amdgcn_wmma_f32_16x16x4_f32(false, a, false, b, (short)0, c,
                                               false, false);
}

__global__ void ncc_zero_kernel(double* gacc) { gacc[0] = 0.0; }

__global__ void __launch_bounds__(32 * WAVES_PER_BLOCK)
ncc_tile_kernel(const float* __restrict__ I, const float* __restrict__ J,
                double* __restrict__ gacc) {
  __shared__ float ldsI[WAVES_PER_BLOCK][32 * TSTRIDE];
  __shared__ float ldsJ[WAVES_PER_BLOCK][32 * TSTRIDE];
  __shared__ float ldsH[WAVES_PER_BLOCK][24 * HSTRIDE];

  const int lane = threadIdx.x & 31;
  const int wave = threadIdx.x >> 5;
  const int tile = blockIdx.x * WAVES_PER_BLOCK + wave;  // grid covers exactly

  const int img = tile / TILES_PER_IMG;
  const int rem = tile % TILES_PER_IMG;
  const int ty = rem / TILES_X;
  const int tx = rem % TILES_X;
  const int r0 = ty * 16;
  const int c0 = tx * 16;

  const float* __restrict__ Ib = I + (size_t)img * (H_IMG * (size_t)W_IMG);
  const float* __restrict__ Jb = J + (size_t)img * (H_IMG * (size_t)W_IMG);

  float* TI = ldsI[wave];
  float* TJ = ldsJ[wave];
  float* HB = ldsH[wave];

  // ---- Stage 24x24 halo tile into LDS (rows 24..31 zeroed for WMMA A blk1) ----
  {
    const int r = lane;  // one row per lane
    float* tiRow = TI + r * TSTRIDE;
    float* tjRow = TJ + r * TSTRIDE;
    const int g = r0 - 4 + r;
    const bool rowok = (r < 24) && ((unsigned)g < (unsigned)H_IMG);
    if (rowok && (c0 >= 16) && (c0 <= W_IMG - 32)) {
      // interior: (c0-4)*4 bytes is 16B aligned since c0 % 16 == 0
      const float4* pI =
          reinterpret_cast<const float4*>(Ib + (size_t)g * W_IMG + (c0 - 4));
      const float4* pJ =
          reinterpret_cast<const float4*>(Jb + (size_t)g * W_IMG + (c0 - 4));
#pragma unroll
      for (int t = 0; t < 6; ++t) {
        float4 a = pI[t];
        float4 b = pJ[t];
        tiRow[4 * t + 0] = a.x; tiRow[4 * t + 1] = a.y;
        tiRow[4 * t + 2] = a.z; tiRow[4 * t + 3] = a.w;
        tjRow[4 * t + 0] = b.x; tjRow[4 * t + 1] = b.y;
        tjRow[4 * t + 2] = b.z; tjRow[4 * t + 3] = b.w;
      }
    } else {
#pragma unroll
      for (int c = 0; c < 24; ++c) {
        const int gc = c0 - 4 + c;
        const bool ok = rowok && ((unsigned)gc < (unsigned)W_IMG);
        tiRow[c] = ok ? Ib[(size_t)g * W_IMG + gc] : 0.0f;
        tjRow[c] = ok ? Jb[(size_t)g * W_IMG + gc] : 0.0f;
      }
    }
  }

  // f32 WMMA 16x16x4 operand mapping (wave32):
  //   A: lane L -> M = L%16, K = kc + 2*(L/16) + v   (v = VGPR 0/1)
  //   B: lane L -> N = L%16, K = kc + 2*(L/16) + v
  //   C/D: lane L, VGPR v -> row = v + 8*(L/16), col = L%16
  const int kgrp = (lane >> 4) << 1;  // 0 or 2
  const int mn = lane & 15;           // A-row / B-col / D-col
  const int drb = (lane >> 4) << 3;   // D row offset: 0 or 8

  v8f sums[5];  // box sums of I, J, I*I, J*J, I*J (16x16 each, D layout)

#pragma unroll
  for (int q = 0; q < 5; ++q) {
    // ---- horizontal pass: H(24x16) = Tq(24x24) x Sband(24x16), 2 row blocks
#pragma unroll
    for (int blk = 0; blk < 2; ++blk) {
      const int mrow = blk * 16;
      v8f acc = {0.f, 0.f, 0.f, 0.f, 0.f, 0.f, 0.f, 0.f};
#pragma unroll
      for (int kc = 0; kc < 24; kc += 4) {
        const int k0 = kc + kgrp;
        const int row = mrow + mn;
        const float i0 = TI[row * TSTRIDE + k0];
        const float i1 = TI[row * TSTRIDE + k0 + 1];
        const float j0 = TJ[row * TSTRIDE + k0];
        const float j1 = TJ[row * TSTRIDE + k0 + 1];
        v2f a;
        if (q == 0)      { a.x = i0;      a.y = i1;      }
        else if (q == 1) { a.x = j0;      a.y = j1;      }
        else if (q == 2) { a.x = i0 * i0; a.y = i1 * i1; }
        else if (q == 3) { a.x = j0 * j0; a.y = j1 * j1; }
        else             { a.x = i0 * j0; a.y = i1 * j1; }
        v2f b;  // S[k][n] = 1 iff n <= k <= n+8
        b.x = (mn <= k0 && k0 <= mn + 8) ? 1.0f : 0.0f;
        b.y = (mn <= k0 + 1 && k0 + 1 <= mn + 8) ? 1.0f : 0.0f;
        acc = wmma_f32(a, b, acc);
      }
#pragma unroll
      for (int v = 0; v < 8; ++v) {
        const int hr = mrow + drb + v;
        if (hr < 24) HB[hr * HSTRIDE + mn] = acc[v];
      }
    }
    // ---- vertical pass: V(16x16) = S'(16x24) x H(24x16)
    {
      v8f acc = {0.f, 0.f, 0.f, 0.f, 0.f, 0.f, 0.f, 0.f};
#pragma unroll
      for (int kc = 0; kc < 24; kc += 4) {
        const int k0 = kc + kgrp;
        v2f a;  // S'[i][r] = 1 iff i <= r <= i+8 (i = mn)
        a.x = (mn <= k0 && k0 <= mn + 8) ? 1.0f : 0.0f;
        a.y = (mn <= k0 + 1 && k0 + 1 <= mn + 8) ? 1.0f : 0.0f;
        v2f b;
        b.x = HB[k0 * HSTRIDE + mn];
        b.y = HB[(k0 + 1) * HSTRIDE + mn];
        acc = wmma_f32(a, b, acc);
      }
      sums[q] = acc;
    }
  }

  // ---- per-pixel cc and local sum (8 pixels per lane, D layout) ----
  float total = 0.0f;
#pragma unroll
  for (int v = 0; v < 8; ++v) {
    const float Is = sums[0][v];
    const float Js = sums[1][v];
    const float I2 = sums[2][v];
    const float J2 = sums[3][v];
    const float IJ = sums[4][v];
    const float inv = 1.0f / 81.0f;
    const float uI = Is * inv;
    const float uJ = Js * inv;
    const float cross = IJ - uJ * Is - uI * Js + uI * uJ * 81.0f;
    const float Iv = I2 - 2.0f * uI * Is + uI * uI * 81.0f;
    const float Jv = J2 - 2.0f * uJ * Js + uJ * uJ * 81.0f;
    const float cc = (cross * cross) / (Iv * Jv + 1e-5f);
    total += cc;
  }

  // ---- wave32 reduction, then one f64 atomic per wave ----
#pragma unroll
  for (int off = 16; off > 0; off >>= 1) total += __shfl_down(total, off, 32);
  if (lane == 0) atomicAdd(gacc, (double)total);
}

__global__ void ncc_finalize_kernel(const double* __restrict__ gacc,
                                    float* __restrict__ out) {
  out[0] = (float)(gacc[0] / (double)(N_IMG * (size_t)H_IMG * W_IMG));
}

extern "C" void kernel_launch(void* const* d_in, const int* in_sizes, int n_in,
                              void* d_out, int out_size, void* d_ws,
                              size_t ws_size, hipStream_t stream) {
  (void)in_sizes; (void)n_in; (void)out_size; (void)ws_size;
  const float* I = (const float*)d_in[0];
  const float* J = (const float*)d_in[1];
  float* out = (float*)d_out;
  double* accum = (double*)d_ws;

  ncc_zero_kernel<<<1, 1, 0, stream>>>(accum);
  ncc_tile_kernel<<<TOTAL_TILES / WAVES_PER_BLOCK, 32 * WAVES_PER_BLOCK, 0,
                    stream>>>(I, J, accum);
  ncc_finalize_kernel<<<1, 1, 0, stream>>>(accum, out);
}